// TopologicalAttention_20194936226612
// MI455X (gfx1250) — compile-verified
//
#include <hip/hip_runtime.h>
#include <hip/hip_bf16.h>

typedef __attribute__((ext_vector_type(2))) float v2f;
typedef __attribute__((ext_vector_type(4))) float v4f;
typedef __attribute__((ext_vector_type(8))) float v8f;

#define BATCH 8
#define NSEQ  1024
#define DM    512
#define DA    64

// ---------------------------------------------------------------------------
// Kernel 1: fused QKV projection.  512 blocks (one per 16-row tile of x),
// 12 waves/block: wave = (mat in {Q,K,V}) x (4 column tiles of 16).
// x tile staged in LDS (row stride 516 floats -> conflict-free WMMA A reads).
// ---------------------------------------------------------------------------
__global__ __launch_bounds__(384) void qkv_kernel(
    const float* __restrict__ x,
    const float* __restrict__ Wq, const float* __restrict__ bq,
    const float* __restrict__ Wk, const float* __restrict__ bk,
    const float* __restrict__ Wv, const float* __restrict__ bv,
    float* __restrict__ Q, float* __restrict__ K, float* __restrict__ V)
{
    __shared__ float xs[16 * 516];
    const int rowbase = blockIdx.x * 16;          // global row in [0, B*N)

    // cooperative load of x tile: 16 x 512 floats as float4
    for (int i = threadIdx.x; i < 16 * 128; i += 384) {
        const int r  = i >> 7;
        const int c4 = i & 127;
        v4f val = *(const v4f*)(x + (size_t)(rowbase + r) * DM + c4 * 4);
        *(v4f*)(&xs[r * 516 + c4 * 4]) = val;
    }
    __syncthreads();

    const int wave = threadIdx.x >> 5;            // 0..11
    const int lane = threadIdx.x & 31;
    const int mat  = wave >> 2;                   // 0=Q 1=K 2=V
    const int ct   = wave & 3;                    // column tile
    const int lr   = lane & 15;
    const int hi   = lane >> 4;                   // half-wave -> K offset +2
    const int colbase = ct * 16;

    const float* W    = (mat == 0) ? Wq : (mat == 1) ? Wk : Wv;
    const float* bias = (mat == 0) ? bq : (mat == 1) ? bk : bv;
    float*       Out  = (mat == 0) ? Q  : (mat == 1) ? K  : V;

    v8f c = {};
    for (int k = 0; k < DM; k += 4) {
        const int ka = k + hi * 2;
        v2f a = *(const v2f*)(&xs[lr * 516 + ka]);          // A: 16x4 tile of x
        v2f b;                                              // B: 4x16 tile of W
        b.x = W[(size_t)ka       * DA + colbase + lr];
        b.y = W[(size_t)(ka + 1) * DA + colbase + lr];
        c = __builtin_amdgcn_wmma_f32_16x16x4_f32(false, a, false, b,
                                                  (short)0, c, false, false);
    }

    const float bb = bias[colbase + lr];
#pragma unroll
    for (int r = 0; r < 8; ++r)
        Out[(size_t)(rowbase + r + hi * 8) * DA + colbase + lr] = c[r] + bb;
}

// ---------------------------------------------------------------------------
// Kernel 2: S = scale * Q K^T + top . Wt + bt      (the 512MB-read kernel)
// One wave per 16x16 tile of S.  Bias dot(16) evaluated per-lane directly in
// the WMMA C layout, with 4x float4 contiguous loads per element.
// `top` is streamed with NON-TEMPORAL loads: it has zero reuse and would
// otherwise evict the L2-resident S / Q / K / V working set (192MB L2).
// ---------------------------------------------------------------------------
__global__ __launch_bounds__(32) void logits_kernel(
    const float* __restrict__ Q, const float* __restrict__ K,
    const float* __restrict__ top, const float* __restrict__ Wt,
    const float* __restrict__ bt, float* __restrict__ S)
{
    const int idx = blockIdx.x;
    const int b   = idx >> 12;            // 64*64 tiles per batch
    const int nt  = (idx >> 6) & 63;
    const int mt  = idx & 63;
    const int lane = threadIdx.x;
    const int lr = lane & 15, hi = lane >> 4;
    const int nbase = nt * 16, mbase = mt * 16;

    const float* Qrow = Q + ((size_t)b * NSEQ + nbase + lr) * DA;  // A rows
    const float* Krow = K + ((size_t)b * NSEQ + mbase + lr) * DA;  // B cols

    v8f c = {};
#pragma unroll
    for (int k = 0; k < DA; k += 4) {
        v2f a  = *(const v2f*)(Qrow + k + hi * 2);
        v2f bb = *(const v2f*)(Krow + k + hi * 2);
        c = __builtin_amdgcn_wmma_f32_16x16x4_f32(false, a, false, bb,
                                                  (short)0, c, false, false);
    }

    float wt[16];
#pragma unroll
    for (int i = 0; i < 16; ++i) wt[i] = Wt[i];   // uniform -> scalar loads
    const float bt0   = bt[0];
    const float scale = 0.125f;                   // 1/sqrt(64)

#pragma unroll
    for (int r = 0; r < 8; ++r) {
        const size_t n = (size_t)nbase + r + hi * 8;
        const float* t = top + (((size_t)b * NSEQ + n) * NSEQ + (mbase + lr)) * 16;
        v4f t0 = __builtin_nontemporal_load((const v4f*)(t));
        v4f t1 = __builtin_nontemporal_load((const v4f*)(t + 4));
        v4f t2 = __builtin_nontemporal_load((const v4f*)(t + 8));
        v4f t3 = __builtin_nontemporal_load((const v4f*)(t + 12));
        float bias = t0.x*wt[0] + t0.y*wt[1] + t0.z*wt[2] + t0.w*wt[3]
                   + t1.x*wt[4] + t1.y*wt[5] + t1.z*wt[6] + t1.w*wt[7]
                   + t2.x*wt[8] + t2.y*wt[9] + t2.z*wt[10]+ t2.w*wt[11]
                   + t3.x*wt[12]+ t3.y*wt[13]+ t3.z*wt[14]+ t3.w*wt[15];
        S[((size_t)b * NSEQ + n) * NSEQ + mbase + lr] = c[r] * scale + bias + bt0;
    }
}

// ---------------------------------------------------------------------------
// Kernel 3: row-wise softmax over N=1024, in place.  One block per row.
// S is L2-resident (32MB in 192MB L2, protected by NT streaming of `top`).
// ---------------------------------------------------------------------------
__global__ __launch_bounds__(256) void softmax_kernel(float* __restrict__ S)
{
    __shared__ float red[256];
    float* p = S + (size_t)blockIdx.x * NSEQ;
    const int t = threadIdx.x;

    v4f v = *(const v4f*)(p + t * 4);
    float m = fmaxf(fmaxf(v.x, v.y), fmaxf(v.z, v.w));
    red[t] = m;
    __syncthreads();
    for (int s = 128; s > 0; s >>= 1) {
        if (t < s) red[t] = fmaxf(red[t], red[t + s]);
        __syncthreads();
    }
    const float rowmax = red[0];
    __syncthreads();

    v.x = __expf(v.x - rowmax);
    v.y = __expf(v.y - rowmax);
    v.z = __expf(v.z - rowmax);
    v.w = __expf(v.w - rowmax);
    red[t] = v.x + v.y + v.z + v.w;
    __syncthreads();
    for (int s = 128; s > 0; s >>= 1) {
        if (t < s) red[t] += red[t + s];
        __syncthreads();
    }
    const float inv = 1.0f / red[0];
    v.x *= inv; v.y *= inv; v.z *= inv; v.w *= inv;
    *(v4f*)(p + t * 4) = v;
}

// ---------------------------------------------------------------------------
// Kernel 4: out = P @ V.  512 blocks (16-row tiles), 4 waves = 4 col tiles.
// P tile (16x1024) staged in LDS with 1028-float row pad (conflict-free).
// Output written non-temporally (never re-read on device).
// ---------------------------------------------------------------------------
__global__ __launch_bounds__(128) void pv_kernel(
    const float* __restrict__ S, const float* __restrict__ V,
    float* __restrict__ out)
{
    __shared__ float ps[16 * 1028];
    const int rowbase = blockIdx.x * 16;          // global row in [0, B*N)
    const int batch   = rowbase >> 10;

    const float* Srow = S + (size_t)rowbase * NSEQ;
    for (int i = threadIdx.x; i < 16 * 256; i += 128) {
        const int r  = i >> 8;
        const int c4 = i & 255;
        v4f val = *(const v4f*)(Srow + (size_t)r * NSEQ + c4 * 4);
        *(v4f*)(&ps[r * 1028 + c4 * 4]) = val;
    }
    __syncthreads();

    const int wave = threadIdx.x >> 5;
    const int lane = threadIdx.x & 31;
    const int lr = lane & 15, hi = lane >> 4;
    const int colbase = wave * 16;
    const float* Vb = V + (size_t)batch * NSEQ * DA;

    v8f c = {};
    for (int k = 0; k < NSEQ; k += 4) {
        const int ka = k + hi * 2;
        v2f a = *(const v2f*)(&ps[lr * 1028 + ka]);
        v2f b;
        b.x = Vb[(size_t)ka       * DA + colbase + lr];
        b.y = Vb[(size_t)(ka + 1) * DA + colbase + lr];
        c = __builtin_amdgcn_wmma_f32_16x16x4_f32(false, a, false, b,
                                                  (short)0, c, false, false);
    }
#pragma unroll
    for (int r = 0; r < 8; ++r)
        __builtin_nontemporal_store(
            c[r], out + (size_t)(rowbase + r + hi * 8) * DA + colbase + lr);
}

// ---------------------------------------------------------------------------
extern "C" void kernel_launch(void* const* d_in, const int* in_sizes, int n_in,
                              void* d_out, int out_size, void* d_ws, size_t ws_size,
                              hipStream_t stream)
{
    (void)in_sizes; (void)n_in; (void)out_size; (void)ws_size;

    const float* x   = (const float*)d_in[0];
    const float* top = (const float*)d_in[1];
    const float* Wq  = (const float*)d_in[2];
    const float* bq  = (const float*)d_in[3];
    const float* Wk  = (const float*)d_in[4];
    const float* bk  = (const float*)d_in[5];
    const float* Wv  = (const float*)d_in[6];
    const float* bv  = (const float*)d_in[7];
    const float* Wt  = (const float*)d_in[8];
    const float* bt  = (const float*)d_in[9];

    float* Q = (float*)d_ws;                          // 8*1024*64
    float* K = Q + (size_t)BATCH * NSEQ * DA;
    float* V = K + (size_t)BATCH * NSEQ * DA;
    float* S = V + (size_t)BATCH * NSEQ * DA;         // 8*1024*1024

    float* out = (float*)d_out;

    qkv_kernel<<<(BATCH * NSEQ) / 16, 384, 0, stream>>>(x, Wq, bq, Wk, bk, Wv, bv,
                                                        Q, K, V);
    logits_kernel<<<BATCH * 64 * 64, 32, 0, stream>>>(Q, K, top, Wt, bt, S);
    softmax_kernel<<<BATCH * NSEQ, 256, 0, stream>>>(S);
    pv_kernel<<<(BATCH * NSEQ) / 16, 128, 0, stream>>>(S, V, out);
}